// IWCrossEntropy_20512763806283
// MI455X (gfx1250) — compile-verified
//
#include <hip/hip_runtime.h>

// ---------------------------------------------------------------------------
// IWCrossEntropy for MI455X (gfx1250, wave32).
// Memory-bound streaming op: ~352 MB traffic -> ~15us floor at 23.3 TB/s.
// Main kernel: coalesced float4 (B128) loads, 19 channels held in VGPRs,
// single-pass max + sum-exp, LDS weight table, wave32 shuffle reductions.
// Final reduction uses V_WMMA_F32_16X16X4_F32 (precision-safe fp32 matrix op).
// ---------------------------------------------------------------------------

#define IGNORE_INDEX 255

constexpr int N_ = 8, C_ = 19, H_ = 512, W_ = 1024;
constexpr int HW_ = H_ * W_;                 // 524288 = 2^19
constexpr int HW_SHIFT = 19;
constexpr int P_ = N_ * HW_;                 // 4194304 pixels
constexpr int QUADS = P_ / 4;                // 1048576 float4 quads
constexpr int TPB = 256;                     // 8 waves per block (wave32)
constexpr int MAIN_BLOCKS = QUADS / TPB;     // 4096 (exact)
constexpr int HIST_BLOCKS = 1024;
constexpr int NC = N_ * C_;                  // 152 histogram bins

typedef __attribute__((ext_vector_type(2))) float v2f;
typedef __attribute__((ext_vector_type(8))) float v8f;

// ---------------- workspace layout (bytes) ----------------
// [0,    608)   : u32 hist[NC]
// [1024, 1632)  : f32 weight[NC]
// [2048, 18432) : f32 psum[MAIN_BLOCKS]
// [18432,34816) : f32 pcnt[MAIN_BLOCKS]

__global__ void iw_init(unsigned int* __restrict__ hist) {
    int t = threadIdx.x;
    if (t < NC) hist[t] = 0u;
}

__global__ void __launch_bounds__(TPB)
iw_hist(const int* __restrict__ tgt, unsigned int* __restrict__ ghist) {
    __shared__ unsigned int bins[NC];
    int tid = threadIdx.x;
    if (tid < NC) bins[tid] = 0u;
    __syncthreads();

    int stride = gridDim.x * blockDim.x;
    for (int q = blockIdx.x * blockDim.x + tid; q < QUADS; q += stride) {
        int p0 = q << 2;
        int n  = p0 >> HW_SHIFT;                       // all 4 pixels share n
        int4 t4 = reinterpret_cast<const int4*>(tgt)[q];
        int tv0 = t4.x, tv1 = t4.y, tv2 = t4.z, tv3 = t4.w;
        if (tv0 >= 0 && tv0 < C_) atomicAdd(&bins[n * C_ + tv0], 1u);
        if (tv1 >= 0 && tv1 < C_) atomicAdd(&bins[n * C_ + tv1], 1u);
        if (tv2 >= 0 && tv2 < C_) atomicAdd(&bins[n * C_ + tv2], 1u);
        if (tv3 >= 0 && tv3 < C_) atomicAdd(&bins[n * C_ + tv3], 1u);
    }
    __syncthreads();
    if (tid < NC) {
        unsigned int v = bins[tid];
        if (v) atomicAdd(&ghist[tid], v);
    }
}

__global__ void iw_weight(const unsigned int* __restrict__ ghist,
                          float* __restrict__ weight) {
    int t = threadIdx.x;
    if (t >= NC) return;
    int n = t / C_;
    float rowsum = 0.0f;
#pragma unroll
    for (int c = 0; c < C_; ++c) {
        unsigned int h = ghist[n * C_ + c];
        rowsum += (h == 0u) ? 1.0f : (float)h;         // hist[hist==0]=1 BEFORE sum
    }
    unsigned int ht = ghist[t];
    float hf = (ht == 0u) ? 1.0f : (float)ht;
    weight[t] = rowsum / hf;                           // RATIO == 1.0
}

__global__ void __launch_bounds__(TPB)
iw_main(const float* __restrict__ x, const int* __restrict__ tgt,
        const float* __restrict__ weight,
        float* __restrict__ psum, float* __restrict__ pcnt) {
    __shared__ float lw[C_];
    int tid = threadIdx.x;
    int q   = blockIdx.x * TPB + tid;                  // exact cover, no tail
    int p0  = q << 2;
    int n   = p0 >> HW_SHIFT;                          // uniform across block
    int hw  = p0 & (HW_ - 1);

    if (tid < C_) lw[tid] = weight[n * C_ + tid];
    __syncthreads();

    // 19 channels x 4 pixels held in VGPRs; loads are lane-contiguous B128.
    const float4* xb = reinterpret_cast<const float4*>(x)
                     + ((size_t)n * C_) * (HW_ / 4) + (hw >> 2);
    float4 v[C_];
#pragma unroll
    for (int c = 0; c < C_; ++c) v[c] = xb[(size_t)c * (HW_ / 4)];

    float m0 = v[0].x, m1 = v[0].y, m2 = v[0].z, m3 = v[0].w;
#pragma unroll
    for (int c = 1; c < C_; ++c) {
        m0 = fmaxf(m0, v[c].x); m1 = fmaxf(m1, v[c].y);
        m2 = fmaxf(m2, v[c].z); m3 = fmaxf(m3, v[c].w);
    }

    int4 t4 = reinterpret_cast<const int4*>(tgt)[q];
    int t0 = t4.x, t1 = t4.y, t2 = t4.z, t3 = t4.w;

    float s0 = 0.f, s1 = 0.f, s2 = 0.f, s3 = 0.f;
    float g0 = 0.f, g1 = 0.f, g2 = 0.f, g3 = 0.f;      // x[target]
#pragma unroll
    for (int c = 0; c < C_; ++c) {
        s0 += __expf(v[c].x - m0); if (c == t0) g0 = v[c].x;
        s1 += __expf(v[c].y - m1); if (c == t1) g1 = v[c].y;
        s2 += __expf(v[c].z - m2); if (c == t2) g2 = v[c].z;
        s3 += __expf(v[c].w - m3); if (c == t3) g3 = v[c].w;
    }
    float lse0 = m0 + __logf(s0), lse1 = m1 + __logf(s1);
    float lse2 = m2 + __logf(s2), lse3 = m3 + __logf(s3);

    bool ok0 = (t0 != IGNORE_INDEX), ok1 = (t1 != IGNORE_INDEX);
    bool ok2 = (t2 != IGNORE_INDEX), ok3 = (t3 != IGNORE_INDEX);
    float w0 = lw[ok0 ? t0 : 0], w1 = lw[ok1 ? t1 : 0];
    float w2 = lw[ok2 ? t2 : 0], w3 = lw[ok3 ? t3 : 0];

    float sum = 0.f, cnt = 0.f;
    if (ok0) { sum += w0 * (lse0 - g0); cnt += 1.f; }
    if (ok1) { sum += w1 * (lse1 - g1); cnt += 1.f; }
    if (ok2) { sum += w2 * (lse2 - g2); cnt += 1.f; }
    if (ok3) { sum += w3 * (lse3 - g3); cnt += 1.f; }

    // wave32 shuffle reduction, then cross-wave via LDS (deterministic order)
#pragma unroll
    for (int off = 16; off > 0; off >>= 1) {
        sum += __shfl_xor(sum, off, 32);
        cnt += __shfl_xor(cnt, off, 32);
    }
    __shared__ float wsum[TPB / 32], wcnt[TPB / 32];
    int lane = tid & 31, wid = tid >> 5;
    if (lane == 0) { wsum[wid] = sum; wcnt[wid] = cnt; }
    __syncthreads();
    if (tid == 0) {
        float S = 0.f, K = 0.f;
#pragma unroll
        for (int i = 0; i < TPB / 32; ++i) { S += wsum[i]; K += wcnt[i]; }
        psum[blockIdx.x] = S;
        pcnt[blockIdx.x] = K;
    }
}

// Final reduction: one wave32. Sums go through V_WMMA_F32_16X16X4_F32 with an
// all-ones B matrix: D = A(16x4) * ones(4x16) + C accumulates 64 values/issue,
// every column of C ends up equal to the per-row-block totals.
__global__ void __launch_bounds__(32)
iw_reduce(const float* __restrict__ psum, const float* __restrict__ pcnt,
          float* __restrict__ out) {
    int lane = threadIdx.x;                            // 0..31, all active

    // valid-pixel count: plain shuffle reduction (uniform trip count)
    float c = 0.f;
    for (int i = lane; i < MAIN_BLOCKS; i += 32) c += pcnt[i];
#pragma unroll
    for (int off = 16; off > 0; off >>= 1) c += __shfl_xor(c, off, 32);

    // A-matrix 16x4 f32 layout: lanes 0-15 {K=0,K=1}, lanes 16-31 {K=2,K=3}
    int M  = lane & 15;
    int K0 = (lane >> 4) << 1;
    v8f acc = {};
    v2f ones; ones[0] = 1.0f; ones[1] = 1.0f;          // B = 4x16 all-ones
    for (int base = 0; base < MAIN_BLOCKS; base += 64) {
        const float* p = psum + base + M * 4 + K0;
        v2f a; a[0] = p[0]; a[1] = p[1];
        acc = __builtin_amdgcn_wmma_f32_16x16x4_f32(
            /*neg_a=*/false, a, /*neg_b=*/false, ones,
            /*c_mod=*/(short)0, acc, /*reuse_a=*/false, /*reuse_b=*/false);
    }
    // lane l (l<16): sum over M=0..7 of column l; lane 16+l: M=8..15.
    float s = acc[0] + acc[1] + acc[2] + acc[3] + acc[4] + acc[5] + acc[6] + acc[7];
    s += __shfl_xor(s, 16, 32);                        // total sum, all lanes

    if (lane == 0) out[0] = s / fmaxf(c, 1.0f);
}

extern "C" void kernel_launch(void* const* d_in, const int* in_sizes, int n_in,
                              void* d_out, int out_size, void* d_ws, size_t ws_size,
                              hipStream_t stream) {
    (void)in_sizes; (void)n_in; (void)out_size; (void)ws_size;
    const float* x   = (const float*)d_in[0];   // [N,C,H,W] fp32
    const int*   tgt = (const int*)d_in[1];     // [N,H,W] int32
    float*       out = (float*)d_out;           // scalar fp32

    char* ws = (char*)d_ws;
    unsigned int* hist   = (unsigned int*)(ws + 0);
    float*        weight = (float*)(ws + 1024);
    float*        psum   = (float*)(ws + 2048);
    float*        pcnt   = (float*)(ws + 2048 + (size_t)MAIN_BLOCKS * 4);

    iw_init  <<<1,            256, 0, stream>>>(hist);
    iw_hist  <<<HIST_BLOCKS,  TPB, 0, stream>>>(tgt, hist);
    iw_weight<<<1,            256, 0, stream>>>(hist, weight);
    iw_main  <<<MAIN_BLOCKS,  TPB, 0, stream>>>(x, tgt, weight, psum, pcnt);
    iw_reduce<<<1,            32,  0, stream>>>(psum, pcnt, out);
}